// RCOptimModel_31894427140436
// MI455X (gfx1250) — compile-verified
//
#include <hip/hip_runtime.h>
#include <math.h>

#define NN 302         // state dim
#define NP 304         // padded to multiple of 16
#define MMC 301        // cols of B input
#define NROOMS 300
#define KNOTS 2001
#define TSTEPS 60000
#define DTF 30.0f
#define SBLK 128       // steps per time-block
#define NBLK 469       // ceil(59999/128)
#define NBLKP 480      // padded to 30*16 column tiles
#define RT 19          // row tiles (304/16)
#define KT 76          // k tiles of 4 (304/4)
#define MSZ (NP*NP)
#define H8 (DTF/8.0f)
#define YSTR 305       // LDS stride (gcd(49,64)=1 -> conflict-free)

typedef float v2f __attribute__((ext_vector_type(2)));
typedef float v8f __attribute__((ext_vector_type(8)));

__device__ __forceinline__ v8f wmma4(v2f a, v2f b, v8f c) {
  // D = A(16x4) * B(4x16) + C(16x16), all f32
  return __builtin_amdgcn_wmma_f32_16x16x4_f32(false, a, false, b, (short)0, c, false, false);
}

__device__ __forceinline__ float sigmoidf(float x) { return 1.0f / (1.0f + __expf(-x)); }

// ---------------- setup kernels ----------------

__global__ void k_build_w(const float* __restrict__ params, const float* __restrict__ lo,
                          const float* __restrict__ hi, float* __restrict__ Wb) {
  int idx = blockIdx.x * blockDim.x + threadIdx.x;
  if (idx >= MSZ) return;
  int r = idx / NP, c = idx % NP;
  float v = 0.f;
  if (r < NN && c < NN) {
    int p = r * NN + c;
    float x = sigmoidf(params[p]);
    v = lo[p] + (hi[p] - lo[p]) * x;
  }
  Wb[idx] = v;
}

__global__ void k_build_q(const float* __restrict__ params, const float* __restrict__ lo,
                          const float* __restrict__ hi, float* __restrict__ Qv) {
  int i = blockIdx.x * blockDim.x + threadIdx.x;
  if (i >= NP) return;
  float v = 0.f;
  if (i < NROOMS) {
    int p = NN * NN + i;
    float x = sigmoidf(params[p]);
    v = lo[p] + (hi[p] - lo[p]) * x;
  }
  Qv[i] = v;
}

// in place: W -> h*(W - diag(rowsum|W|))  == h*A
__global__ void k_rows(float* Wb) {
  int r = threadIdx.x;
  if (r >= NN) return;
  float s = 0.f;
  for (int c = 0; c < NN; ++c) s += fabsf(Wb[r * NP + c]);
  for (int c = 0; c < NN; ++c) {
    float v = Wb[r * NP + c] - (c == r ? s : 0.f);
    Wb[r * NP + c] = DTF * v;
  }
}

// ---------------- generic 304x304x304 WMMA GEMM: C = A @ B ----------------
__global__ void k_gemm(const float* __restrict__ A, const float* __restrict__ B,
                       float* __restrict__ C) {
  int tb = blockIdx.x;
  int ti = tb / RT, tj = tb % RT;
  int lane = threadIdx.x & 31;
  int rowl = lane & 15;
  int hl = lane >> 4;          // half-wave select
  int r0 = ti * 16, c0 = tj * 16;
  v8f acc = {0.f, 0.f, 0.f, 0.f, 0.f, 0.f, 0.f, 0.f};
  for (int kt = 0; kt < KT; ++kt) {
    int kb = kt * 4 + hl * 2;
    v2f a = *(const v2f*)(A + (r0 + rowl) * NP + kb);   // A-operand lane layout
    v2f b;
    b.x = B[kb * NP + c0 + rowl];
    b.y = B[(kb + 1) * NP + c0 + rowl];
    acc = wmma4(a, b, acc);
  }
  for (int v = 0; v < 8; ++v) {
    int row = r0 + v + hl * 8;   // C/D layout: VGPR v -> row v (lanes<16) / v+8
    C[row * NP + c0 + rowl] = acc[v];
  }
}

// M = I + hA + (hA)^2/2 + (hA)^3/6 + (hA)^4/24
__global__ void k_combine(const float* Bh, const float* B2, const float* B3, const float* B4,
                          float* M) {
  int idx = blockIdx.x * blockDim.x + threadIdx.x;
  if (idx >= MSZ) return;
  int r = idx / NP, c = idx % NP;
  M[idx] = (r == c ? 1.f : 0.f) + Bh[idx] + 0.5f * B2[idx] + (1.f / 6.f) * B3[idx]
           + (1.f / 24.f) * B4[idx];
}

// repack M into WMMA A-operand tile layout for coalesced inner-loop loads
__global__ void k_repack(const float* __restrict__ M, float* __restrict__ At) {
  int idx = blockIdx.x * blockDim.x + threadIdx.x;
  if (idx >= MSZ) return;
  int t = idx >> 6;            // tile id (rt*KT + kt), 64 floats per tile
  int q = idx & 63;
  int lane = q >> 1, g = q & 1;
  int rt = t / KT, kt = t % KT;
  int row = rt * 16 + (lane & 15);
  int k = kt * 4 + ((lane >> 4) << 1) + g;
  At[idx] = M[row * NP + k];
}

// ---------------- forcing vectors: v1..v4 = N_i b0, w = (Sum N_i) c ----------------
__global__ void k_vectors(const float* __restrict__ Bmat, const float* __restrict__ Bh,
                          const float* __restrict__ Qv, const float* __restrict__ iv,
                          float* wv, float* v1, float* v2, float* v3, float* v4,
                          float* Ycp, float* out) {
  __shared__ float cur[NP];
  int r = threadIdx.x;
  float b0 = 0.f, cc = 0.f;
  if (r < NN) {
    b0 = Bmat[r * MMC];
    float s = 0.f;
    for (int m = 0; m < NROOMS; ++m) s += Bmat[r * MMC + 1 + m] * Qv[m];
    cc = s;
  }
  // powers of (hA) applied to b0
  float p0 = b0, p1 = 0.f, p2 = 0.f, p3 = 0.f;
  if (r < NP) cur[r] = (r < NN) ? b0 : 0.f;
  __syncthreads();
  if (r < NN) { float s = 0; for (int k = 0; k < NP; ++k) s += Bh[r * NP + k] * cur[k]; p1 = s; }
  __syncthreads();
  if (r < NP) cur[r] = (r < NN) ? p1 : 0.f;
  __syncthreads();
  if (r < NN) { float s = 0; for (int k = 0; k < NP; ++k) s += Bh[r * NP + k] * cur[k]; p2 = s; }
  __syncthreads();
  if (r < NP) cur[r] = (r < NN) ? p2 : 0.f;
  __syncthreads();
  if (r < NN) { float s = 0; for (int k = 0; k < NP; ++k) s += Bh[r * NP + k] * cur[k]; p3 = s; }
  __syncthreads();
  if (r < NP) {
    float a1 = 0, a2 = 0, a3 = 0, a4 = 0;
    if (r < NN) {
      a1 = H8 * (p0 + p1 + (1.f / 3.f) * (p2 + p3));
      a2 = H8 * (3.f * p0 + 2.f * p1 + p2);
      a3 = H8 * (3.f * p0 + p1);
      a4 = H8 * p0;
    }
    v1[r] = a1; v2[r] = a2; v3[r] = a3; v4[r] = a4;
  }
  // powers applied to c
  float q0 = cc, q1 = 0.f, q2 = 0.f, q3 = 0.f;
  if (r < NP) cur[r] = (r < NN) ? cc : 0.f;
  __syncthreads();
  if (r < NN) { float s = 0; for (int k = 0; k < NP; ++k) s += Bh[r * NP + k] * cur[k]; q1 = s; }
  __syncthreads();
  if (r < NP) cur[r] = (r < NN) ? q1 : 0.f;
  __syncthreads();
  if (r < NN) { float s = 0; for (int k = 0; k < NP; ++k) s += Bh[r * NP + k] * cur[k]; q2 = s; }
  __syncthreads();
  if (r < NP) cur[r] = (r < NN) ? q2 : 0.f;
  __syncthreads();
  if (r < NN) { float s = 0; for (int k = 0; k < NP; ++k) s += Bh[r * NP + k] * cur[k]; q3 = s; }
  if (r < NP) {
    float wr = 0.f;
    if (r < NN) wr = H8 * (8.f * q0 + 4.f * q1 + (4.f / 3.f) * q2 + (1.f / 3.f) * q3);
    wv[r] = wr;
    float y0 = (r < NN) ? iv[r] : 0.f;
    Ycp[r] = y0;                 // checkpoint 0 = iv
    if (r < NN) out[r] = y0;     // output row 0 = iv
  }
}

// ---------------- sequential checkpoint scan with M^128 ----------------
__global__ void k_scan(const float* __restrict__ Mp, const float* __restrict__ Cagg,
                       float* __restrict__ Ycp) {
  __shared__ float y[NP];
  int r = threadIdx.x;
  if (r < NP) y[r] = Ycp[r];
  __syncthreads();
  for (int b = 0; b < NBLK; ++b) {
    float s = 0.f;
    if (r < NP) {
      for (int k = 0; k < NP; ++k) s += Mp[r * NP + k] * y[k];
      s += Cagg[b * NP + r];
    }
    __syncthreads();
    if (r < NP) { y[r] = s; Ycp[(b + 1) * NP + r] = s; }
    __syncthreads();
  }
}

// ---------------- batched block recurrence: Y <- M Y + g  (mode 0: aggregate, 1: fill) ----------------
__global__ void k_block(int mode, const float* __restrict__ At,
                        const float* __restrict__ wv, const float* __restrict__ v1,
                        const float* __restrict__ v2, const float* __restrict__ v3,
                        const float* __restrict__ v4, const float* __restrict__ tout_t,
                        const float* __restrict__ tout_vals, const int* __restrict__ t0p,
                        const float* __restrict__ Ycp, float* __restrict__ Cagg,
                        float* __restrict__ out) {
  __shared__ float Yl[16 * YSTR];
  __shared__ float Gl[16 * YSTR];
  __shared__ float taus[64];
  const int wgBase = blockIdx.x * 16;   // 16 time-blocks (columns) per workgroup
  const int tid = threadIdx.x;
  const int wave = tid >> 5, lane = tid & 31;
  const int colc = lane & 15, hl = lane >> 4;
  const float t0f = (float)(*t0p);
  const float tt0 = tout_t[0];
  const float invdk = 1.0f / (tout_t[1] - tout_t[0]);

  for (int idx = tid; idx < 16 * NP; idx += 256) {
    int c = idx / NP, row = idx - c * NP;
    float v = 0.f;
    if (mode == 1) v = Ycp[(wgBase + c) * NP + row];
    Yl[c * YSTR + row] = v;
  }
  __syncthreads();

  int rts[3]; int nrt = 0;
  for (int rt = wave; rt < RT; rt += 8) rts[nrt++] = rt;

  for (int l = 0; l < SBLK; ++l) {
    // 4 interpolated stage temperatures per column
    if (tid < 64) {
      int c = tid >> 2, si = tid & 3;
      const float sfr[4] = {0.f, 1.f / 3.f, 2.f / 3.f, 1.f};
      float n = (float)((wgBase + c) * SBLK + l);
      float t = (n + sfr[si]) * DTF + t0f;
      float u = (t - tt0) * invdk;
      u = fminf(fmaxf(u, 0.f), (float)(KNOTS - 1));
      int i0 = (int)u; if (i0 > KNOTS - 2) i0 = KNOTS - 2;
      float f = u - (float)i0;
      float a = tout_vals[i0];
      taus[tid] = a + f * (tout_vals[i0 + 1] - a);
    }
    __syncthreads();
    // g_n = w + sum_i tau_i v_i
    for (int idx = tid; idx < 16 * NP; idx += 256) {
      int c = idx / NP, row = idx - c * NP;
      float g = 0.f;
      if (row < NN)
        g = wv[row] + taus[c * 4 + 0] * v1[row] + taus[c * 4 + 1] * v2[row]
                    + taus[c * 4 + 2] * v3[row] + taus[c * 4 + 3] * v4[row];
      Gl[c * YSTR + row] = g;
    }
    __syncthreads();

    // Ynew = M @ Y + G  via f32 WMMA, row tiles split across 8 waves
    v8f acc[3];
    for (int i = 0; i < nrt; ++i) {
      int rt = rts[i];
      v8f a8;
      for (int v = 0; v < 8; ++v) a8[v] = Gl[colc * YSTR + rt * 16 + v + hl * 8];
      const float* Ap = At + (rt * KT) * 64 + lane * 2;
      for (int kt = 0; kt < KT; ++kt) {
        v2f a = *(const v2f*)(Ap + kt * 64);    // pre-repacked A-operand, coalesced
        int kb = kt * 4 + hl * 2;
        v2f b;
        b.x = Yl[colc * YSTR + kb];
        b.y = Yl[colc * YSTR + kb + 1];
        a8 = wmma4(a, b, a8);
      }
      acc[i] = a8;
    }
    __syncthreads();
    for (int i = 0; i < nrt; ++i) {
      int rt = rts[i];
      for (int v = 0; v < 8; ++v) {
        int row = rt * 16 + v + hl * 8;
        float val = acc[i][v];
        Yl[colc * YSTR + row] = val;
        if (mode == 1 && row < NN) {
          int nout = (wgBase + colc) * SBLK + l + 1;
          if (nout < TSTEPS) out[nout * NN + row] = val;
        }
      }
    }
    __syncthreads();
  }

  if (mode == 0) {
    for (int idx = tid; idx < 16 * NP; idx += 256) {
      int c = idx / NP, row = idx - c * NP;
      Cagg[(wgBase + c) * NP + row] = Yl[c * YSTR + row];
    }
  }
}

extern "C" void kernel_launch(void* const* d_in, const int* in_sizes, int n_in,
                              void* d_out, int out_size, void* d_ws, size_t ws_size,
                              hipStream_t stream) {
  (void)in_sizes; (void)n_in; (void)out_size; (void)ws_size;
  const float* params = (const float*)d_in[0];
  const float* lo     = (const float*)d_in[1];
  const float* hi     = (const float*)d_in[2];
  const float* Bm     = (const float*)d_in[3];
  const float* toutt  = (const float*)d_in[4];
  const float* toutv  = (const float*)d_in[5];
  const float* iv     = (const float*)d_in[6];
  /* d_in[7] = t_eval: uniform grid with h == DT, encoded in constants */
  const int*   t0p    = (const int*)d_in[8];
  float* out = (float*)d_out;
  float* ws = (float*)d_ws;

  float* Bh = ws;                  // h*A
  float* B2 = ws + 1 * (size_t)MSZ;
  float* B3 = ws + 2 * (size_t)MSZ;
  float* B4 = ws + 3 * (size_t)MSZ;
  float* Mm = ws + 4 * (size_t)MSZ;
  float* Pa = ws + 5 * (size_t)MSZ;
  float* Pb = ws + 6 * (size_t)MSZ;
  float* At = ws + 7 * (size_t)MSZ;
  float* Qv = ws + 8 * (size_t)MSZ;
  float* wv = Qv + NP;
  float* v1 = wv + NP; float* v2 = v1 + NP; float* v3 = v2 + NP; float* v4 = v3 + NP;
  float* Cagg = v4 + NP;
  float* Ycp  = Cagg + (size_t)NBLKP * NP;

  k_build_w<<<(MSZ + 255) / 256, 256, 0, stream>>>(params, lo, hi, Bh);
  k_build_q<<<2, 256, 0, stream>>>(params, lo, hi, Qv);
  k_rows<<<1, 320, 0, stream>>>(Bh);

  dim3 gt(RT * RT);
  k_gemm<<<gt, 32, 0, stream>>>(Bh, Bh, B2);
  k_gemm<<<gt, 32, 0, stream>>>(B2, Bh, B3);
  k_gemm<<<gt, 32, 0, stream>>>(B2, B2, B4);
  k_combine<<<(MSZ + 255) / 256, 256, 0, stream>>>(Bh, B2, B3, B4, Mm);
  k_repack<<<(MSZ + 255) / 256, 256, 0, stream>>>(Mm, At);

  // M^128 by repeated squaring
  k_gemm<<<gt, 32, 0, stream>>>(Mm, Mm, Pa);  // 2
  k_gemm<<<gt, 32, 0, stream>>>(Pa, Pa, Pb);  // 4
  k_gemm<<<gt, 32, 0, stream>>>(Pb, Pb, Pa);  // 8
  k_gemm<<<gt, 32, 0, stream>>>(Pa, Pa, Pb);  // 16
  k_gemm<<<gt, 32, 0, stream>>>(Pb, Pb, Pa);  // 32
  k_gemm<<<gt, 32, 0, stream>>>(Pa, Pa, Pb);  // 64
  k_gemm<<<gt, 32, 0, stream>>>(Pb, Pb, Pa);  // 128 -> Pa

  k_vectors<<<1, 320, 0, stream>>>(Bm, Bh, Qv, iv, wv, v1, v2, v3, v4, Ycp, out);

  // P1: per-block forcing aggregates
  k_block<<<NBLKP / 16, 256, 0, stream>>>(0, At, wv, v1, v2, v3, v4, toutt, toutv, t0p,
                                          Ycp, Cagg, out);
  // P2: checkpoint scan
  k_scan<<<1, 320, 0, stream>>>(Pa, Cagg, Ycp);
  // P3: parallel fill of all 60000 states
  k_block<<<NBLKP / 16, 256, 0, stream>>>(1, At, wv, v1, v2, v3, v4, toutt, toutv, t0p,
                                          Ycp, Cagg, out);
}